// AGCLoss_forward_70411693851274
// MI455X (gfx1250) — compile-verified
//
#include <hip/hip_runtime.h>
#include <hip/hip_bf16.h>
#include <math.h>

// ---------------- problem constants ----------------
#define PP      6400      // samples per image (80*80) = P1 = P2
#define KC      49        // patch channels (7*7)
#define KPAD    52        // padded K (13 * 4)
#define KSTEPS  13        // KPAD / 4  (V_WMMA_F32_16X16X4_F32 k-steps)
#define NG      169       // patch grid size: (512-7)/3+1
#define IMGW    512
#define EPSF    2.2204460492503131e-16f
#define NTILES  400       // PP / 16 column tiles
#define WAVES   8         // waves per workgroup (256 threads, wave32)
#define LAMBDA_OCC 0.05f

typedef float v2f __attribute__((ext_vector_type(2)));
typedef float v8f __attribute__((ext_vector_type(8)));

static __device__ inline v8f wmma4(v2f a, v2f b, v8f c) {
  // D = A(16x4 f32) * B(4x16 f32) + C(16x16 f32), full-rate f32 WMMA on CDNA5
  return __builtin_amdgcn_wmma_f32_16x16x4_f32(false, a, false, b, (short)0, c,
                                               false, false);
}

// ---------------- stage 1: gather sampled patch features ----------------
// X[c][s] = target_img[(iy*3+c/7)*512 + ix*3+c%7], pad rows c>=49 zeroed.
__global__ __launch_bounds__(256) void k_gather(
    const float* __restrict__ timg, const float* __restrict__ rimg,
    const float* __restrict__ tfld, const float* __restrict__ rfld,
    float* __restrict__ X, float* __restrict__ Y)
{
  int gid = blockIdx.x * 256 + threadIdx.x;
  const int total = 2 * KPAD * PP;
  if (gid >= total) return;
  int which = gid / (KPAD * PP);
  int rem   = gid - which * (KPAD * PP);
  int c     = rem / PP;
  int s     = rem - c * PP;
  const float* img = which ? rimg : timg;
  const float* fld = which ? rfld : tfld;
  float*       dst = which ? Y : X;
  float v = 0.0f;
  if (c < KC) {
    float gx = fld[2 * s + 0];
    float gy = fld[2 * s + 1];
    // jnp.round(f*169 - 0.5) with round-half-to-even == rintf (default RNE)
    int ix = (int)rintf(gx * (float)NG - 0.5f);
    int iy = (int)rintf(gy * (float)NG - 0.5f);
    ix = min(max(ix, 0), NG - 1);
    iy = min(max(iy, 0), NG - 1);
    int py = c / 7, px = c - py * 7;
    v = img[(iy * 3 + py) * IMGW + (ix * 3 + px)];
  }
  dst[c * PP + s] = v;
}

// ---------------- stage 2: refer-feature channel means ----------------
__global__ __launch_bounds__(256) void k_ymean(const float* __restrict__ Y,
                                               float* __restrict__ ymean)
{
  __shared__ float red[256];
  int c = blockIdx.x;  // 0..48
  float s = 0.0f;
  for (int i = threadIdx.x; i < PP; i += 256) s += Y[c * PP + i];
  red[threadIdx.x] = s;
  __syncthreads();
  for (int st = 128; st > 0; st >>= 1) {
    if (threadIdx.x < st) red[threadIdx.x] += red[threadIdx.x + st];
    __syncthreads();
  }
  if (threadIdx.x == 0) ymean[c] = red[0] / (float)PP;
}

// ---------------- stage 3: in-place L2 normalize per sample ----------------
__global__ __launch_bounds__(256) void k_normalize(float* __restrict__ X,
                                                   float* __restrict__ Y,
                                                   const float* __restrict__ ymean)
{
  int s = blockIdx.x * 256 + threadIdx.x;
  if (s >= PP) return;
  float sx = 0.0f, sy = 0.0f;
  for (int c = 0; c < KC; ++c) {
    float m  = ymean[c];
    float dx = X[c * PP + s] - m;
    float dy = Y[c * PP + s] - m;
    sx += dx * dx;
    sy += dy * dy;
  }
  float ivx = 1.0f / (sqrtf(sx) + EPSF);
  float ivy = 1.0f / (sqrtf(sy) + EPSF);
  for (int c = 0; c < KC; ++c) {
    float m = ymean[c];
    X[c * PP + s] = (X[c * PP + s] - m) * ivx;
    Y[c * PP + s] = (Y[c * PP + s] - m) * ivy;
  }
}

// ---------------- fragment loader (ISA 16x4 f32 A-layout) ----------------
// lanes 0-15: regs hold K = {k0, k0+1}; lanes 16-31: K = {k0+2, k0+3}; col = base + (lane&15)
static __device__ inline void load_frag13(const float* __restrict__ M, int base,
                                          int half, int t, v2f f[KSTEPS])
{
#pragma unroll
  for (int ks = 0; ks < KSTEPS; ++ks) {
    int k0 = 4 * ks + 2 * half;
    f[ks].x = M[(k0 + 0) * PP + base + t];
    f[ks].y = M[(k0 + 1) * PP + base + t];
  }
}

// ---------------- pass 1: per-row argmin of d ----------------
__global__ __launch_bounds__(256) void k_pass1_argmin(
    const float* __restrict__ X, const float* __restrict__ Y,
    int* __restrict__ argidx)
{
  __shared__ float smin[16][128];
  __shared__ int   sidx[16][128];
  const int p0 = blockIdx.x * 16;
  const int tid = threadIdx.x;
  const int wave = tid >> 5, lane = tid & 31, half = lane >> 4, t = lane & 15;

  v2f af[KSTEPS];
  load_frag13(X, p0, half, t, af);

  float rmin[8];
  int   ridx[8];
#pragma unroll
  for (int i = 0; i < 8; ++i) { rmin[i] = 3.4e38f; ridx[i] = 0; }

  for (int jt = wave; jt < NTILES; jt += WAVES) {
    int q0 = jt * 16;
    v2f bf[KSTEPS];
    load_frag13(Y, q0, half, t, bf);
    v8f acc = {0.f, 0.f, 0.f, 0.f, 0.f, 0.f, 0.f, 0.f};
#pragma unroll
    for (int ks = 0; ks < KSTEPS; ++ks) acc = wmma4(af[ks], bf[ks], acc);
    int q = q0 + t;
#pragma unroll
    for (int i = 0; i < 8; ++i) {
      float d = fmaxf((1.0f - acc[i]) * 0.5f, 0.0f);
      if (d < rmin[i]) { rmin[i] = d; ridx[i] = q; }  // strict < keeps earliest q
    }
  }
#pragma unroll
  for (int i = 0; i < 8; ++i) {
    int m = 8 * half + i;
    smin[m][wave * 16 + t] = rmin[i];
    sidx[m][wave * 16 + t] = ridx[i];
  }
  __syncthreads();
  if (tid < 16) {
    float best = 3.4e38f;
    int bidx = 0x7fffffff;
    for (int j = 0; j < 128; ++j) {
      float v = smin[tid][j];
      int   ii = sidx[tid][j];
      if (v < best || (v == best && ii < bidx)) { best = v; bidx = ii; }
    }
    argidx[p0 + tid] = bidx;
  }
}

// ---------------- histogram ----------------
__global__ __launch_bounds__(256) void k_zero_counts(int* __restrict__ counts)
{
  int i = blockIdx.x * 256 + threadIdx.x;
  if (i < PP) counts[i] = 0;
}

__global__ __launch_bounds__(256) void k_hist(const int* __restrict__ argidx,
                                              int* __restrict__ counts)
{
  int p = blockIdx.x * 256 + threadIdx.x;
  if (p < PP) atomicAdd(&counts[argidx[p]], 1);
}

// ---------------- pass 2: per-row min of d' = d + 0.05*counts[q] ----------------
__global__ __launch_bounds__(256) void k_pass2_dmin(
    const float* __restrict__ X, const float* __restrict__ Y,
    const int* __restrict__ counts, float* __restrict__ dmin)
{
  __shared__ float smin[16][128];
  const int p0 = blockIdx.x * 16;
  const int tid = threadIdx.x;
  const int wave = tid >> 5, lane = tid & 31, half = lane >> 4, t = lane & 15;

  v2f af[KSTEPS];
  load_frag13(X, p0, half, t, af);

  float rmin[8];
#pragma unroll
  for (int i = 0; i < 8; ++i) rmin[i] = 3.4e38f;

  for (int jt = wave; jt < NTILES; jt += WAVES) {
    int q0 = jt * 16;
    v2f bf[KSTEPS];
    load_frag13(Y, q0, half, t, bf);
    v8f acc = {0.f, 0.f, 0.f, 0.f, 0.f, 0.f, 0.f, 0.f};
#pragma unroll
    for (int ks = 0; ks < KSTEPS; ++ks) acc = wmma4(af[ks], bf[ks], acc);
    float occ = LAMBDA_OCC * (float)counts[q0 + t];
#pragma unroll
    for (int i = 0; i < 8; ++i) {
      float d = fmaxf((1.0f - acc[i]) * 0.5f, 0.0f) + occ;
      rmin[i] = fminf(rmin[i], d);
    }
  }
#pragma unroll
  for (int i = 0; i < 8; ++i) smin[8 * half + i][wave * 16 + t] = rmin[i];
  __syncthreads();
  if (tid < 16) {
    float best = 3.4e38f;
    for (int j = 0; j < 128; ++j) best = fminf(best, smin[tid][j]);
    dmin[p0 + tid] = best;
  }
}

// ---------------- pass 3: per-row sum of exp weights + row loss ----------------
__global__ __launch_bounds__(256) void k_pass3_sum(
    const float* __restrict__ X, const float* __restrict__ Y,
    const int* __restrict__ counts, const float* __restrict__ dmin,
    float* __restrict__ rowloss)
{
  __shared__ float ssum[16][128];
  const int p0 = blockIdx.x * 16;
  const int tid = threadIdx.x;
  const int wave = tid >> 5, lane = tid & 31, half = lane >> 4, t = lane & 15;

  v2f af[KSTEPS];
  load_frag13(X, p0, half, t, af);

  float coef[8], rsum[8];
#pragma unroll
  for (int i = 0; i < 8; ++i) {
    float dm = dmin[p0 + 8 * half + i];
    coef[i] = 2.0f / (dm + EPSF);  // e = 2 - d * coef
    rsum[i] = 0.0f;
  }

  for (int jt = wave; jt < NTILES; jt += WAVES) {
    int q0 = jt * 16;
    v2f bf[KSTEPS];
    load_frag13(Y, q0, half, t, bf);
    v8f acc = {0.f, 0.f, 0.f, 0.f, 0.f, 0.f, 0.f, 0.f};
#pragma unroll
    for (int ks = 0; ks < KSTEPS; ++ks) acc = wmma4(af[ks], bf[ks], acc);
    float occ = LAMBDA_OCC * (float)counts[q0 + t];
#pragma unroll
    for (int i = 0; i < 8; ++i) {
      float d = fmaxf((1.0f - acc[i]) * 0.5f, 0.0f) + occ;
      rsum[i] += expf(2.0f - d * coef[i]);
    }
  }
#pragma unroll
  for (int i = 0; i < 8; ++i) ssum[8 * half + i][wave * 16 + t] = rsum[i];
  __syncthreads();
  if (tid < 16) {
    float total = 0.0f;
    for (int j = 0; j < 128; ++j) total += ssum[tid][j];
    float dm   = dmin[p0 + tid];
    float emax = 2.0f - dm * (2.0f / (dm + EPSF));  // log(max w)
    // -log(CX) = log(sum w) - log(max w)
    rowloss[p0 + tid] = logf(total) - emax;
  }
}

// ---------------- finalize: mean over rows ----------------
__global__ __launch_bounds__(256) void k_finalize(const float* __restrict__ rowloss,
                                                  float* __restrict__ out)
{
  __shared__ float red[256];
  float s = 0.0f;
  for (int i = threadIdx.x; i < PP; i += 256) s += rowloss[i];
  red[threadIdx.x] = s;
  __syncthreads();
  for (int st = 128; st > 0; st >>= 1) {
    if (threadIdx.x < st) red[threadIdx.x] += red[threadIdx.x + st];
    __syncthreads();
  }
  if (threadIdx.x == 0) out[0] = red[0] / (float)PP;
}

// ---------------- host-side launcher ----------------
extern "C" void kernel_launch(void* const* d_in, const int* in_sizes, int n_in,
                              void* d_out, int out_size, void* d_ws, size_t ws_size,
                              hipStream_t stream) {
  (void)in_sizes; (void)n_in; (void)out_size; (void)ws_size;
  const float* timg = (const float*)d_in[0];  // [1,1,512,512]
  const float* rimg = (const float*)d_in[1];  // [1,1,512,512]
  const float* tfld = (const float*)d_in[2];  // [1,80,80,2]
  const float* rfld = (const float*)d_in[3];  // [1,80,80,2]
  float* out = (float*)d_out;                 // scalar f32

  // workspace layout (~2.8 MB)
  float* X       = (float*)d_ws;              // KPAD*PP
  float* Y       = X + KPAD * PP;             // KPAD*PP
  float* ymean   = Y + KPAD * PP;             // 64
  int*   counts  = (int*)(ymean + 64);        // PP
  int*   argidx  = counts + PP;               // PP
  float* dmin    = (float*)(argidx + PP);     // PP
  float* rowloss = dmin + PP;                 // PP

  // 1) gather both feature matrices (pad K rows to 52 with zeros)
  {
    int total = 2 * KPAD * PP;
    k_gather<<<(total + 255) / 256, 256, 0, stream>>>(timg, rimg, tfld, rfld, X, Y);
  }
  // 2) refer channel means
  k_ymean<<<KC, 256, 0, stream>>>(Y, ymean);
  // 3) normalize in place
  k_normalize<<<(PP + 255) / 256, 256, 0, stream>>>(X, Y, ymean);
  // 4) occurrence histogram prep
  k_zero_counts<<<(PP + 255) / 256, 256, 0, stream>>>(counts);
  // 5) GEMM pass 1: per-row argmin of d
  k_pass1_argmin<<<NTILES, 256, 0, stream>>>(X, Y, argidx);
  // 6) histogram of nearest matches
  k_hist<<<(PP + 255) / 256, 256, 0, stream>>>(argidx, counts);
  // 7) GEMM pass 2: per-row min of reweighted d
  k_pass2_dmin<<<NTILES, 256, 0, stream>>>(X, Y, counts, dmin);
  // 8) GEMM pass 3: per-row exp-weight sums -> row losses
  k_pass3_sum<<<NTILES, 256, 0, stream>>>(X, Y, counts, dmin, rowloss);
  // 9) scalar loss
  k_finalize<<<1, 256, 0, stream>>>(rowloss, out);
}